// GeneralizedGraphDiffusion_82703890251803
// MI455X (gfx1250) — compile-verified
//
#include <hip/hip_runtime.h>

typedef __attribute__((ext_vector_type(2))) float v2f;
typedef __attribute__((ext_vector_type(4))) float v4f;
typedef __attribute__((ext_vector_type(8))) float v8f;

#define Nn 8192
#define Dd 256
#define KC 128

__global__ __launch_bounds__(256) void ggd_fused_kernel(
    const float* __restrict__ theta,
    const float* __restrict__ T,      // [4, N, N]
    const float* __restrict__ x,      // [N, D]
    const float* __restrict__ a,      // [N, N]
    const float* __restrict__ W,      // [D, D]
    const float* __restrict__ b,      // [D]
    const float* __restrict__ alpha,  // [D]
    float* __restrict__ out)          // [N, D]
{
    // LDS: fused-A chunk (32 x 128, stride 132) + out1 row-strip (32 x 256, stride 260)
    __shared__ float ldsA[32 * 132];   // 16896 B
    __shared__ float ldsO[32 * 260];   // 33280 B

    const int t    = threadIdx.x;      // 0..255
    const int wave = t >> 5;           // 0..7
    const int lane = t & 31;
    const int l16  = lane & 15;
    const int kh   = lane >> 4;        // lane half: picks K offset {0,2} in fragments
    const int row0 = blockIdx.x * 32;  // this WG's output rows [row0, row0+32)
    const int c0   = wave * 32 + l16;  // this lane's column in the wave's first tile

    // cooperative loader mapping: 8 threads per row, 16 floats per thread
    const int lr = t >> 3;             // 0..31
    const int lc = (t & 7) * 16;       // 0,16,...,112

    const float th0 = theta[0], th1 = theta[1], th2 = theta[2], th3 = theta[3];
    const long long NN = (long long)Nn * Nn;

    v8f acc00 = {}, acc01 = {}, acc10 = {}, acc11 = {};

    // Pipeline registers: staged raw streams (next chunk) + fused tile (current chunk)
    v4f st0[4], st1[4], st2[4], st3[4], sta[4];
    v4f f[4];

    // Stage the 5 read-once streams non-temporally (don't pollute L2: x must stay hot)
    #define STAGE(K0)                                                              \
    {                                                                              \
        const long long sb = (long long)(row0 + lr) * Nn + (K0) + lc;              \
        _Pragma("unroll")                                                          \
        for (int c4 = 0; c4 < 4; ++c4) {                                           \
            const long long off = sb + c4 * 4;                                     \
            st0[c4] = __builtin_nontemporal_load((const v4f*)(T + off));           \
            st1[c4] = __builtin_nontemporal_load((const v4f*)(T + NN + off));      \
            st2[c4] = __builtin_nontemporal_load((const v4f*)(T + 2 * NN + off));  \
            st3[c4] = __builtin_nontemporal_load((const v4f*)(T + 3 * NN + off));  \
            sta[c4] = __builtin_nontemporal_load((const v4f*)(a + off));           \
        }                                                                          \
    }

    #define FUSE()                                                                 \
        _Pragma("unroll")                                                          \
        for (int c4 = 0; c4 < 4; ++c4)                                             \
            f[c4] = (th0 * st0[c4] + th1 * st1[c4] +                               \
                     th2 * st2[c4] + th3 * st3[c4]) * sta[c4];

    // Prologue: stage + fuse chunk 0
    STAGE(0)
    FUSE()

    for (int k0 = 0; k0 < Nn; k0 += KC) {
        // ---- publish fused A tile for this chunk
        #pragma unroll
        for (int c4 = 0; c4 < 4; ++c4)
            *(v4f*)&ldsA[lr * 132 + lc + c4 * 4] = f[c4];
        __syncthreads();

        // issue next chunk's stream loads NOW, unconditionally (clamped address on the
        // last chunk so there is no branch between the barrier and the WMMA loop and
        // the scheduler cannot sink the loads); waits land at FUSE after the WMMAs.
        const int kn = (k0 + KC < Nn) ? (k0 + KC) : 0;
        STAGE(kn)
        const int kp = (k0 + 2 * KC < Nn) ? (k0 + 2 * KC) : 0;
        {
            const long long pb = (long long)(row0 + lr) * Nn + kp + lc;
            __builtin_prefetch(T + pb, 0, 0);
            __builtin_prefetch(T + NN + pb, 0, 0);
            __builtin_prefetch(T + 2 * NN + pb, 0, 0);
            __builtin_prefetch(T + 3 * NN + pb, 0, 0);
            __builtin_prefetch(a + pb, 0, 0);
        }

        // ---- WMMA over the chunk: 2 row-blocks x 2 col-tiles per wave
        #pragma unroll 4
        for (int kk = 0; kk < KC; kk += 4) {
            v2f fa0 = *(const v2f*)&ldsA[l16 * 132 + kk + kh * 2];
            v2f fa1 = *(const v2f*)&ldsA[(16 + l16) * 132 + kk + kh * 2];
            const float* xp = x + (long long)(k0 + kk + kh * 2) * Dd;
            v2f fb0 = { xp[c0],      xp[Dd + c0]      };
            v2f fb1 = { xp[c0 + 16], xp[Dd + c0 + 16] };
            acc00 = __builtin_amdgcn_wmma_f32_16x16x4_f32(false, fa0, false, fb0, (short)0, acc00, false, false);
            acc01 = __builtin_amdgcn_wmma_f32_16x16x4_f32(false, fa0, false, fb1, (short)0, acc01, false, false);
            acc10 = __builtin_amdgcn_wmma_f32_16x16x4_f32(false, fa1, false, fb0, (short)0, acc10, false, false);
            acc11 = __builtin_amdgcn_wmma_f32_16x16x4_f32(false, fa1, false, fb1, (short)0, acc11, false, false);
        }

        // fuse the staged loads for the next chunk (this is where load waits land)
        FUSE()
        __syncthreads();
    }

    // ---- PReLU, stage out1 strip in LDS
    const float al0 = alpha[c0];
    const float al1 = alpha[c0 + 16];
    #pragma unroll
    for (int j = 0; j < 8; ++j) {
        const int rr = kh * 8 + j;   // C/D layout: lanes 16-31 hold rows M=8..15
        float v;
        v = acc00[j]; ldsO[rr * 260 + c0]             = (v >= 0.f) ? v : al0 * v;
        v = acc01[j]; ldsO[rr * 260 + c0 + 16]        = (v >= 0.f) ? v : al1 * v;
        v = acc10[j]; ldsO[(16 + rr) * 260 + c0]      = (v >= 0.f) ? v : al0 * v;
        v = acc11[j]; ldsO[(16 + rr) * 260 + c0 + 16] = (v >= 0.f) ? v : al1 * v;
    }
    __syncthreads();

    // ---- second GEMM: out2 = prelu(out1) @ W^T + b   (B[k][n] = W[n][k])
    v8f d00 = {}, d01 = {}, d10 = {}, d11 = {};
    #pragma unroll 4
    for (int kk = 0; kk < Dd; kk += 4) {
        v2f fa0 = *(const v2f*)&ldsO[l16 * 260 + kk + kh * 2];
        v2f fa1 = *(const v2f*)&ldsO[(16 + l16) * 260 + kk + kh * 2];
        v2f fb0 = *(const v2f*)&W[c0 * Dd + kk + kh * 2];
        v2f fb1 = *(const v2f*)&W[(c0 + 16) * Dd + kk + kh * 2];
        d00 = __builtin_amdgcn_wmma_f32_16x16x4_f32(false, fa0, false, fb0, (short)0, d00, false, false);
        d01 = __builtin_amdgcn_wmma_f32_16x16x4_f32(false, fa0, false, fb1, (short)0, d01, false, false);
        d10 = __builtin_amdgcn_wmma_f32_16x16x4_f32(false, fa1, false, fb0, (short)0, d10, false, false);
        d11 = __builtin_amdgcn_wmma_f32_16x16x4_f32(false, fa1, false, fb1, (short)0, d11, false, false);
    }

    // ---- bias + store
    const float bb0 = b[c0], bb1 = b[c0 + 16];
    #pragma unroll
    for (int j = 0; j < 8; ++j) {
        const int rr = kh * 8 + j;
        out[(long long)(row0 + rr) * Dd + c0]           = d00[j] + bb0;
        out[(long long)(row0 + rr) * Dd + c0 + 16]      = d01[j] + bb1;
        out[(long long)(row0 + 16 + rr) * Dd + c0]      = d10[j] + bb0;
        out[(long long)(row0 + 16 + rr) * Dd + c0 + 16] = d11[j] + bb1;
    }

    #undef STAGE
    #undef FUSE
}

extern "C" void kernel_launch(void* const* d_in, const int* in_sizes, int n_in,
                              void* d_out, int out_size, void* d_ws, size_t ws_size,
                              hipStream_t stream) {
    (void)in_sizes; (void)n_in; (void)d_ws; (void)ws_size; (void)out_size;
    const float* theta = (const float*)d_in[0];
    const float* T     = (const float*)d_in[1];
    const float* x     = (const float*)d_in[2];
    const float* a     = (const float*)d_in[3];
    const float* W     = (const float*)d_in[4];
    const float* b     = (const float*)d_in[5];
    const float* alpha = (const float*)d_in[6];
    float* out = (float*)d_out;

    dim3 grid(Nn / 32);
    dim3 block(256);
    ggd_fused_kernel<<<grid, block, 0, stream>>>(theta, T, x, a, W, b, alpha, out);
}